// HeteroGNN_55602646614065
// MI455X (gfx1250) — compile-verified
//
#include <hip/hip_runtime.h>

#define N_MOV  200000
#define N_CREW 100000
#define N_CAST 100000
#define NE     600000
#define HDIM   128
#define OUTD   16

typedef float v2f __attribute__((ext_vector_type(2)));
typedef float v8f __attribute__((ext_vector_type(8)));

// ---------------- utility kernels ----------------

__global__ void zero_k(float* __restrict__ p, long n) {
  long i = (long)blockIdx.x * blockDim.x + threadIdx.x;
  if (i < n) p[i] = 0.0f;
}

__global__ void addmat_k(const float* __restrict__ a, const float* __restrict__ b,
                         float* __restrict__ o, int n) {
  int i = blockIdx.x * blockDim.x + threadIdx.x;
  if (i < n) o[i] = a[i] + b[i];
}

__global__ void deg_k(const int* __restrict__ dst, float* __restrict__ deg, int E) {
  int i = blockIdx.x * blockDim.x + threadIdx.x;
  if (i < E) atomicAdd(&deg[dst[i]], 1.0f);
}

// agg[dst[e], :] += x[src[e], :]  (D = 1<<LOGD features; 4 features per thread)
template<int LOGD>
__global__ void scatter4_k(const float* __restrict__ x, const int* __restrict__ src,
                           const int* __restrict__ dst, float* __restrict__ agg, long total4) {
  long i = (long)blockIdx.x * blockDim.x + threadIdx.x;
  if (i < total4) {
    constexpr int QD = LOGD - 2;                 // quads per row
    int e = (int)(i >> QD);
    int f = (int)((i & ((1 << QD) - 1)) << 2);   // feature base
    const float4 v = *(const float4*)&x[((size_t)src[e] << LOGD) + f];
    float* p = &agg[((size_t)dst[e] << LOGD) + f];
    atomicAdd(p + 0, v.x);
    atomicAdd(p + 1, v.y);
    atomicAdd(p + 2, v.z);
    atomicAdd(p + 3, v.w);
  }
}

// ---------------- WMMA fp32 building block ----------------
// One K-chunk (64) of A[rows 16]@W accumulated into acc, staged via LDS.
// Lane layouts for V_WMMA_F32_16X16X4_F32:
//   A: lane l holds A[m=l%16][k = k0 + v + 2h]  (v=0,1; h = l>=16)
//   B: lane l holds B[k = k0 + v + 2h][n = col]
//   C: vgpr r, lane l -> M = r + 8h, N = col
#define KC 64
#define KP (KC + 1)
#define NPAD (HDIM + 2)

__device__ __forceinline__ void wmma_chunk(const float* __restrict__ lA,
                                           const float* __restrict__ lW,
                                           int n16, int h, int col, v8f& acc) {
  const float* pa = &lA[n16 * KP];
  const float* pb = &lW[col];
#pragma unroll
  for (int k0 = 0; k0 < KC; k0 += 4) {
    v2f a, b;
    a.x = pa[k0 + 2 * h + 0];
    a.y = pa[k0 + 2 * h + 1];
    b.x = pb[(size_t)(k0 + 2 * h + 0) * NPAD];
    b.y = pb[(size_t)(k0 + 2 * h + 1) * NPAD];
    acc = __builtin_amdgcn_wmma_f32_16x16x4_f32(false, a, false, b, (short)0, acc,
                                                false, false);
  }
}

// ---------------- fused SAGE node update ----------------
// out[M,128] = relu( alpha * ( sum_t  diag(1/max(deg_t,1)) * A_t[M,K_t] @ W_t[K_t,128]
//                              + b1 + b2 ) )
// 256 threads = 8 wave32s; each wave owns one 16x16 tile of the 16x128 block stripe.
__global__ __launch_bounds__(256) void sage_update_k(
    const float* A0, const float* deg0, const float* W0, int K0,
    const float* A1, const float* deg1, const float* W1, int K1,
    const float* A2, const float* deg2, const float* W2, int K2,
    const float* __restrict__ b1, const float* __restrict__ b2,
    float alpha, float* __restrict__ Cout, int M)
{
  __shared__ float lA[16 * KP];
  __shared__ float lW[KC * NPAD];

  const int tid  = threadIdx.x;
  const int row0 = blockIdx.x * 16;
  const int lane = tid & 31;
  const int h    = lane >> 4;
  const int n16  = lane & 15;
  const int nt   = tid >> 5;          // wave id == n-tile
  const int col  = nt * 16 + n16;

  const float* As[3] = {A0, A1, A2};
  const float* Ds[3] = {deg0, deg1, deg2};
  const float* Ws[3] = {W0, W1, W2};
  const int    Ks[3] = {K0, K1, K2};

  v8f acc = {};

#pragma unroll
  for (int t = 0; t < 3; ++t) {
    const float* A = As[t];
    if (!A) continue;
    const float* deg = Ds[t];
    const float* W = Ws[t];
    const int K = Ks[t];
    for (int kc = 0; kc < K; kc += KC) {
      __syncthreads();  // protect previous chunk's LDS reads
      for (int i = tid; i < KC * HDIM; i += 256) {
        int kk = i >> 7, n = i & 127;
        lW[kk * NPAD + n] = W[(size_t)(kc + kk) * HDIM + n];
      }
      for (int i = tid; i < 16 * KC; i += 256) {
        int r = i >> 6, c = i & 63;
        int gr = row0 + r;
        float v = 0.0f;
        if (gr < M) {
          v = A[(size_t)gr * K + kc + c];
          if (deg) v *= 1.0f / fmaxf(deg[gr], 1.0f);
        }
        lA[r * KP + c] = v;
      }
      __syncthreads();
      wmma_chunk(lA, lW, n16, h, col, acc);
    }
  }

  float bias = b1[col] + (b2 ? b2[col] : 0.0f);
#pragma unroll
  for (int r = 0; r < 8; ++r) {
    int grow = row0 + r + 8 * h;
    if (grow < M) {
      float v = alpha * (acc[r] + bias);
      Cout[(size_t)grow * HDIM + col] = fmaxf(v, 0.0f);
    }
  }
}

// ---------------- final projection: out[M,16] = X[M,128] @ W[128,16] + b ----------------
// 256 threads = 8 wave32s; each wave owns one 16-row m-tile of a 128-row stripe.
__global__ __launch_bounds__(256) void proj_k(
    const float* __restrict__ X, const float* __restrict__ W,
    const float* __restrict__ bias, float* __restrict__ Out, int M)
{
  constexpr int NPS = OUTD + 2;
  __shared__ float lA[128 * KP];
  __shared__ float lW[KC * NPS];

  const int tid  = threadIdx.x;
  const int row0 = blockIdx.x * 128;
  const int lane = tid & 31;
  const int h    = lane >> 4;
  const int n16  = lane & 15;
  const int mt   = tid >> 5;
  const int col  = n16;  // single n-tile (N=16)

  v8f acc = {};

  for (int kc = 0; kc < HDIM; kc += KC) {
    __syncthreads();
    for (int i = tid; i < KC * OUTD; i += 256) {
      int kk = i >> 4, n = i & 15;
      lW[kk * NPS + n] = W[(size_t)(kc + kk) * OUTD + n];
    }
    for (int i = tid; i < 128 * KC; i += 256) {
      int r = i >> 6, c = i & 63;
      int gr = row0 + r;
      lA[r * KP + c] = (gr < M) ? X[(size_t)gr * HDIM + kc + c] : 0.0f;
    }
    __syncthreads();

    const float* pa = &lA[(mt * 16 + n16) * KP];
    const float* pb = &lW[col];
#pragma unroll
    for (int k0 = 0; k0 < KC; k0 += 4) {
      v2f a, b;
      a.x = pa[k0 + 2 * h + 0];
      a.y = pa[k0 + 2 * h + 1];
      b.x = pb[(size_t)(k0 + 2 * h + 0) * NPS];
      b.y = pb[(size_t)(k0 + 2 * h + 1) * NPS];
      acc = __builtin_amdgcn_wmma_f32_16x16x4_f32(false, a, false, b, (short)0, acc,
                                                  false, false);
    }
  }

  float bv = bias[col];
#pragma unroll
  for (int r = 0; r < 8; ++r) {
    int grow = row0 + mt * 16 + r + 8 * h;
    if (grow < M) Out[(size_t)grow * OUTD + col] = acc[r] + bv;
  }
}

// ---------------- host orchestration ----------------

static inline int blks(long n, int t) { return (int)((n + t - 1) / t); }

extern "C" void kernel_launch(void* const* d_in, const int* in_sizes, int n_in,
                              void* d_out, int out_size, void* d_ws, size_t ws_size,
                              hipStream_t stream) {
  (void)in_sizes; (void)n_in; (void)out_size; (void)ws_size;

  const float* x_movies = (const float*)d_in[0];
  const float* x_crew   = (const float*)d_in[1];
  const float* x_cast   = (const float*)d_in[2];
  const int*   src_cm   = (const int*)d_in[3];
  const int*   dst_cm   = (const int*)d_in[4];
  const int*   src_am   = (const int*)d_in[5];
  const int*   dst_am   = (const int*)d_in[6];

  const float* Wl_cm_0 = (const float*)d_in[7];
  const float* Wr_cm_0 = (const float*)d_in[8];
  const float* b_cm_0  = (const float*)d_in[9];
  const float* Wl_am_0 = (const float*)d_in[10];
  const float* Wr_am_0 = (const float*)d_in[11];
  const float* b_am_0  = (const float*)d_in[12];
  const float* Wl_mc_0 = (const float*)d_in[13];
  const float* Wr_mc_0 = (const float*)d_in[14];
  const float* b_mc_0  = (const float*)d_in[15];
  const float* Wl_ma_0 = (const float*)d_in[16];
  const float* Wr_ma_0 = (const float*)d_in[17];
  const float* b_ma_0  = (const float*)d_in[18];
  const float* Wl_cm_1 = (const float*)d_in[19];
  const float* Wr_cm_1 = (const float*)d_in[20];
  const float* b_cm_1  = (const float*)d_in[21];
  const float* Wl_am_1 = (const float*)d_in[22];
  const float* Wr_am_1 = (const float*)d_in[23];
  const float* b_am_1  = (const float*)d_in[24];
  const float* lin_W   = (const float*)d_in[31];
  const float* lin_b   = (const float*)d_in[32];

  // workspace layout (floats), total ~411 MB
  float* ws   = (float*)d_ws;
  float* xm1  = ws;                                   // 200000*128
  float* xc1  = xm1 + (size_t)N_MOV * HDIM;           // 100000*128
  float* xa1  = xc1 + (size_t)N_CREW * HDIM;          // 100000*128
  float* xm2  = xc1;  // aliases xc1+xa1 (dead after layer-1 scatters)
  float* aggA = xa1 + (size_t)N_CAST * HDIM;          // 200000*128
  float* aggB = aggA + (size_t)N_MOV * HDIM;          // 200000*128
  float* degA = aggB + (size_t)N_MOV * HDIM;          // 200000
  float* degB = degA + N_MOV;                         // 200000
  float* Wsum = degB + N_MOV;                         // 128*128

  const long e64q  = (long)NE * (64 / 4);
  const long e128q = (long)NE * (128 / 4);

  // ================= Layer 0 =================
  // movies <- crew (cm) and cast (am)
  zero_k<<<blks((long)N_MOV * 64, 256), 256, 0, stream>>>(aggA, (long)N_MOV * 64);
  zero_k<<<blks(N_MOV, 256), 256, 0, stream>>>(degA, N_MOV);
  deg_k<<<blks(NE, 256), 256, 0, stream>>>(dst_cm, degA, NE);
  scatter4_k<6><<<blks(e64q, 256), 256, 0, stream>>>(x_crew, src_cm, dst_cm, aggA, e64q);

  zero_k<<<blks((long)N_MOV * 64, 256), 256, 0, stream>>>(aggB, (long)N_MOV * 64);
  zero_k<<<blks(N_MOV, 256), 256, 0, stream>>>(degB, N_MOV);
  deg_k<<<blks(NE, 256), 256, 0, stream>>>(dst_am, degB, NE);
  scatter4_k<6><<<blks(e64q, 256), 256, 0, stream>>>(x_cast, src_am, dst_am, aggB, e64q);

  addmat_k<<<blks(HDIM * HDIM, 256), 256, 0, stream>>>(Wr_cm_0, Wr_am_0, Wsum, HDIM * HDIM);
  sage_update_k<<<N_MOV / 16, 256, 0, stream>>>(
      aggA, degA, Wl_cm_0, 64,
      aggB, degB, Wl_am_0, 64,
      x_movies, nullptr, Wsum, 128,
      b_cm_0, b_am_0, 0.5f, xm1, N_MOV);

  // crew <- movies (mc): src = dst_cm, dst = src_cm
  zero_k<<<blks((long)N_CREW * 128, 256), 256, 0, stream>>>(aggA, (long)N_CREW * 128);
  zero_k<<<blks(N_CREW, 256), 256, 0, stream>>>(degA, N_CREW);
  deg_k<<<blks(NE, 256), 256, 0, stream>>>(src_cm, degA, NE);
  scatter4_k<7><<<blks(e128q, 256), 256, 0, stream>>>(x_movies, dst_cm, src_cm, aggA, e128q);
  sage_update_k<<<N_CREW / 16, 256, 0, stream>>>(
      aggA, degA, Wl_mc_0, 128,
      x_crew, nullptr, Wr_mc_0, 64,
      nullptr, nullptr, nullptr, 0,
      b_mc_0, nullptr, 1.0f, xc1, N_CREW);

  // cast <- movies (ma): src = dst_am, dst = src_am
  zero_k<<<blks((long)N_CAST * 128, 256), 256, 0, stream>>>(aggB, (long)N_CAST * 128);
  zero_k<<<blks(N_CAST, 256), 256, 0, stream>>>(degB, N_CAST);
  deg_k<<<blks(NE, 256), 256, 0, stream>>>(src_am, degB, NE);
  scatter4_k<7><<<blks(e128q, 256), 256, 0, stream>>>(x_movies, dst_am, src_am, aggB, e128q);
  sage_update_k<<<N_CAST / 16, 256, 0, stream>>>(
      aggB, degB, Wl_ma_0, 128,
      x_cast, nullptr, Wr_ma_0, 64,
      nullptr, nullptr, nullptr, 0,
      b_ma_0, nullptr, 1.0f, xa1, N_CAST);

  // ================= Layer 1 (movies only; crew/cast outputs are dead) =================
  zero_k<<<blks((long)N_MOV * 128, 256), 256, 0, stream>>>(aggA, (long)N_MOV * 128);
  zero_k<<<blks(N_MOV, 256), 256, 0, stream>>>(degA, N_MOV);
  deg_k<<<blks(NE, 256), 256, 0, stream>>>(dst_cm, degA, NE);
  scatter4_k<7><<<blks(e128q, 256), 256, 0, stream>>>(xc1, src_cm, dst_cm, aggA, e128q);

  zero_k<<<blks((long)N_MOV * 128, 256), 256, 0, stream>>>(aggB, (long)N_MOV * 128);
  zero_k<<<blks(N_MOV, 256), 256, 0, stream>>>(degB, N_MOV);
  deg_k<<<blks(NE, 256), 256, 0, stream>>>(dst_am, degB, NE);
  scatter4_k<7><<<blks(e128q, 256), 256, 0, stream>>>(xa1, src_am, dst_am, aggB, e128q);

  // xc1/xa1 now dead -> xm2 (aliased region) may be written
  addmat_k<<<blks(HDIM * HDIM, 256), 256, 0, stream>>>(Wr_cm_1, Wr_am_1, Wsum, HDIM * HDIM);
  sage_update_k<<<N_MOV / 16, 256, 0, stream>>>(
      aggA, degA, Wl_cm_1, 128,
      aggB, degB, Wl_am_1, 128,
      xm1, nullptr, Wsum, 128,
      b_cm_1, b_am_1, 0.5f, xm2, N_MOV);

  // ================= Final linear: out = xm2 @ lin_W + lin_b =================
  proj_k<<<(N_MOV + 127) / 128, 256, 0, stream>>>(xm2, lin_W, lin_b, (float*)d_out, N_MOV);
}